// DubonNet_23055384444945
// MI455X (gfx1250) — compile-verified
//
#include <hip/hip_runtime.h>
#include <hip/hip_bf16.h>
#include <math.h>

typedef __attribute__((ext_vector_type(2))) float v2f;
typedef __attribute__((ext_vector_type(8))) float v8f;

#define NPTS   28
#define NPAIR  (NPTS * NPTS)       // 784 pairs
#define NRBF   50
#define NBASIS 8
#define NHID   16
#define MROWS  (NPAIR * NBASIS)    // 6272  rows   (p,b1)
#define NCOLS  (NPAIR * NHID)      // 12544 cols   (q,h)
#define MTILES (MROWS / 16)        // 392
#define NTILES (NCOLS / 16)        // 784 = 49 * 16

__global__ void zero_out_kernel(float* out) { out[0] = 0.f; }

// One thread per pair p=(i,j): builds the K=4-padded GEMM operands.
//   Arow[p*8+b1]  = (u0,u1,u2,0) * sK_p[b1]        (A matrix row, K0..K3)
//   Bcol[p*16+h]  = (u0,u1,u2,0) * g_p[h]          (B matrix col, K0..K3)
__global__ __launch_bounds__(128)
void precompute_kernel(const float* __restrict__ x,
                       const float* __restrict__ means,
                       const float* __restrict__ betas,
                       const float* __restrict__ K,
                       const float* __restrict__ Q,
                       const float* __restrict__ W0,
                       float4* __restrict__ Arow,
                       float4* __restrict__ Bcol)
{
    int p = blockIdx.x * blockDim.x + threadIdx.x;
    if (p >= NPAIR) return;
    int i = p / NPTS, j = p % NPTS;

    float dx0 = x[j * 3 + 0] - x[i * 3 + 0];
    float dx1 = x[j * 3 + 1] - x[i * 3 + 1];
    float dx2 = x[j * 3 + 2] - x[i * 3 + 2];
    float norm = sqrtf(dx0 * dx0 + dx1 * dx1 + dx2 * dx2);
    float inv  = norm > 0.f ? 1.f / norm : 1.f;   // matches jnp.where safe-divide
    float u0 = dx0 * inv, u1 = dx1 * inv, u2 = dx2 * inv;
    float en = __expf(-norm);                     // exp(ALPHA*(CL - norm)), ALPHA=1, CL=0

    float sK[NBASIS], sQ[NBASIS];
#pragma unroll
    for (int b = 0; b < NBASIS; b++) { sK[b] = 0.f; sQ[b] = 0.f; }
    for (int r = 0; r < NRBF; r++) {
        float d = en - means[r];
        float s = __expf(-betas[r] * d * d);
#pragma unroll
        for (int b = 0; b < NBASIS; b++) {
            sK[b] += s * K[r * NBASIS + b];
            sQ[b] += s * Q[r * NBASIS + b];
        }
    }
#pragma unroll
    for (int b = 0; b < NBASIS; b++)
        Arow[p * NBASIS + b] = make_float4(u0 * sK[b], u1 * sK[b], u2 * sK[b], 0.f);

#pragma unroll
    for (int h = 0; h < NHID; h++) {
        float g = 0.f;
#pragma unroll
        for (int b = 0; b < NBASIS; b++) g += sQ[b] * W0[b * NHID + h];
        Bcol[p * NHID + h] = make_float4(u0 * g, u1 * g, u2 * g, 0.f);
    }
}

// tanh(x) = 1 - 2/(exp(2x)+1), with raw v_exp_f32 / v_rcp_f32 (no IEEE div
// expansion). Argument clamped so rcp never sees inf/0 pathologies.
__device__ __forceinline__ float fast_tanh(float v) {
    v = fminf(fmaxf(v, -15.f), 15.f);                        // v_med3_num_f32
    float e = __builtin_amdgcn_exp2f(v * 2.88539008177793f); // 2*log2(e)
    float r = __builtin_amdgcn_rcpf(e + 1.f);                // v_rcp_f32
    return fmaf(-2.f, r, 1.f);
}

// 8 waves/block; block b owns row tile b; wave w strides over column tiles,
// 2 tiles per iteration (49 iterations, no tail). Two B loads are issued
// back-to-back so the first WMMA waits only on loadcnt<=1 and the first tanh
// block hides the second load's latency.
__global__ __launch_bounds__(256)
void gemm_tanh_reduce_kernel(const float2* __restrict__ A2,  // 2x float2 per row (K0K1 | K2K3)
                             const float2* __restrict__ B2,  // 2x float2 per col (K0K1 | K2K3)
                             const float* __restrict__ W1,
                             float* __restrict__ out)
{
    const int lane = threadIdx.x & 31;
    const int wave = threadIdx.x >> 5;       // 0..7
    const int half = lane >> 4;              // 0: K=0,1   1: K=2,3
    const int l15  = lane & 15;
    const int mt   = blockIdx.x;             // row tile 0..391

    // A operand (16x4 f32): lanes 0-15 -> (K0,K1) of row M=l15; lanes 16-31 -> (K2,K3)
    float2 av = A2[2 * (mt * 16 + l15) + half];
    v2f a; a[0] = av.x; a[1] = av.y;

    const float w1 = W1[l15];                // column h == lane&15 in C/D layout
    float acc = 0.f;

    for (int nt = wave; nt < NTILES; nt += 16) {             // nt, nt+8 both valid
        float2 bv0 = B2[2 * (nt * 16 + l15) + half];
        float2 bv1 = B2[2 * ((nt + 8) * 16 + l15) + half];

        v2f b0; b0[0] = bv0.x; b0[1] = bv0.y;
        v2f b1; b1[0] = bv1.x; b1[1] = bv1.y;

        v8f c0 = {}, c1 = {};
        // D = A(16x4) * B(4x16) + 0 -- args: (neg_a, A, neg_b, B, c_mod, C, reuse_a, reuse_b)
        c0 = __builtin_amdgcn_wmma_f32_16x16x4_f32(false, a, false, b0, (short)0, c0,
                                                   false, false);
        float s0 = 0.f;
#pragma unroll
        for (int r = 0; r < 8; r++) s0 += fast_tanh(c0[r]);

        c1 = __builtin_amdgcn_wmma_f32_16x16x4_f32(false, a, false, b1, (short)0, c1,
                                                   false, false);
        float s1 = 0.f;
#pragma unroll
        for (int r = 0; r < 8; r++) s1 += fast_tanh(c1[r]);

        acc += s0 + s1;
    }
    acc *= w1;

    // wave32 butterfly reduction, then one atomic per wave
#pragma unroll
    for (int off = 16; off > 0; off >>= 1) acc += __shfl_xor(acc, off, 32);
    if (lane == 0) atomicAdd(out, acc);
}

extern "C" void kernel_launch(void* const* d_in, const int* in_sizes, int n_in,
                              void* d_out, int out_size, void* d_ws, size_t ws_size,
                              hipStream_t stream) {
    (void)in_sizes; (void)n_in; (void)out_size; (void)ws_size;
    // inputs: h(unused), x, means, betas, K, Q, W0, W1
    const float* x     = (const float*)d_in[1];
    const float* means = (const float*)d_in[2];
    const float* betas = (const float*)d_in[3];
    const float* K     = (const float*)d_in[4];
    const float* Q     = (const float*)d_in[5];
    const float* W0    = (const float*)d_in[6];
    const float* W1    = (const float*)d_in[7];
    float* out = (float*)d_out;

    float4* Arow = (float4*)d_ws;                                           // 6272 * 16B
    float4* Bcol = (float4*)((char*)d_ws + (size_t)MROWS * sizeof(float4)); // 12544 * 16B

    zero_out_kernel<<<1, 1, 0, stream>>>(out);
    precompute_kernel<<<(NPAIR + 127) / 128, 128, 0, stream>>>(
        x, means, betas, K, Q, W0, Arow, Bcol);
    gemm_tanh_reduce_kernel<<<MTILES, 256, 0, stream>>>(
        (const float2*)Arow, (const float2*)Bcol, W1, out);
}